// SO3DiTLayer_67319317397779
// MI455X (gfx1250) — compile-verified
//
#include <hip/hip_runtime.h>
#include <hip/hip_bf16.h>
#include <math.h>

#define N_NODES 6000
#define N_EDGES 72000
#define PDIM 2
#define FDIM 64
#define HHEADS 4
#define TDIM 64
#define NFMLP 256
#define MDIM 9
#define CDIMV 1664
#define ROWF (PDIM*MDIM*FDIM) /* 1152 */
#define EPSV 1e-6f
#define EPB 4            /* edges per block in edge kernels */
#define HSLD 260         /* padded LDS row stride (floats) to avoid bank conflicts */

typedef __attribute__((ext_vector_type(2))) float v2f;
typedef __attribute__((ext_vector_type(8))) float v8f;

__device__ inline v8f wmma4(v2f a, v2f b, v8f c) {
  return __builtin_amdgcn_wmma_f32_16x16x4_f32(false, a, false, b, (short)0, c, false, false);
}

// 16x16 f32 tile: acc += A(16xK, row-major lda) * B(Kx16, row-major ldb)
template<int K>
__device__ inline v8f tile_gemm(const float* __restrict__ A, int lda,
                                const float* __restrict__ B, int ldb,
                                v8f acc, int lo, int hi) {
#pragma unroll
  for (int k = 0; k < K; k += 4) {
    v2f a, b;
    a.x = A[lo*lda + k + 2*hi];
    a.y = A[lo*lda + k + 2*hi + 1];
    b.x = B[(k + 2*hi)*ldb + lo];
    b.y = B[(k + 2*hi + 1)*ldb + lo];
    acc = wmma4(a, b, acc);
  }
  return acc;
}

__device__ inline int deg_of(int m){ return (m==0) ? 0 : ((m<4) ? 1 : 2); }
__device__ inline float siluf(float x){ return x / (1.0f + expf(-x)); }
__device__ inline float geluf(float x){
  return 0.5f*x*(1.0f + tanhf(0.7978845608028654f*(x + 0.044715f*x*x*x)));
}
__device__ inline unsigned f2key(float f){
  unsigned b = __float_as_uint(f);
  return (b & 0x80000000u) ? ~b : (b | 0x80000000u);
}
__device__ inline float key2f(unsigned k){
  return (k & 0x80000000u) ? __uint_as_float(k & 0x7fffffffu) : __uint_as_float(~k);
}

// ---------- K0: u = silu(LN(features_time)) ----------
__global__ void k_ln_silu(const float* __restrict__ ft, const float* __restrict__ sc,
                          const float* __restrict__ bi, float* __restrict__ u){
  __shared__ float red[TDIM];
  int n = blockIdx.x, t = threadIdx.x;
  float x = ft[n*TDIM + t];
  red[t] = x; __syncthreads();
  for (int st=32; st>0; st>>=1){ if (t<st) red[t]+=red[t+st]; __syncthreads(); }
  float mu = red[0] * (1.0f/TDIM); __syncthreads();
  float d = x - mu;
  red[t] = d*d; __syncthreads();
  for (int st=32; st>0; st>>=1){ if (t<st) red[t]+=red[t+st]; __syncthreads(); }
  float var = red[0] * (1.0f/TDIM);
  float y = d * rsqrtf(var + EPSV) * sc[t] + bi[t];
  u[n*TDIM + t] = siluf(y);
}

// ---------- K1: c = u @ W_c + b_c   (6000x64)@(64x1664), WMMA ----------
__global__ void k_gemm_c(const float* __restrict__ u, const float* __restrict__ Wc,
                         const float* __restrict__ bc, float* __restrict__ c){
  int ct = blockIdx.x, nt = blockIdx.y;
  int lane = threadIdx.x, lo = lane & 15, hi = lane >> 4;
  v8f acc = {};
  acc = tile_gemm<TDIM>(u + nt*16*TDIM, TDIM, Wc + ct*16, CDIMV, acc, lo, hi);
  float bv = bc[ct*16 + lo];
#pragma unroll
  for (int r=0;r<8;r++){
    int row = hi*8 + r;
    c[(nt*16+row)*CDIMV + ct*16 + lo] = acc[r] + bv;
  }
}

// ---------- K2: x_pre = modulate(eqv_layernorm(xin), g, b) ----------
__global__ void k_eqv_mod(const float* __restrict__ xin, const float* __restrict__ c,
                          float* __restrict__ xo, int g_off, int b_off){
  __shared__ float red[FDIM];
  int np = blockIdx.x; int n = np / PDIM, p = np % PDIM;
  int f = threadIdx.x;
  int base = n*ROWF + p*MDIM*FDIM;
  float xr[MDIM];
#pragma unroll
  for (int m=0;m<MDIM;m++) xr[m] = xin[base + m*FDIM + f];
  red[f] = xr[0]; __syncthreads();
  for (int st=32; st>0; st>>=1){ if (f<st) red[f]+=red[f+st]; __syncthreads(); }
  float mu = red[0] * (1.0f/FDIM); __syncthreads();
  xr[0] -= mu;
  float n2[3];
  n2[0] = xr[0]*xr[0];
  n2[1] = (xr[1]*xr[1] + xr[2]*xr[2] + xr[3]*xr[3]) * (1.0f/3.0f);
  n2[2] = (xr[4]*xr[4] + xr[5]*xr[5] + xr[6]*xr[6] + xr[7]*xr[7] + xr[8]*xr[8]) * (1.0f/5.0f);
  float rinv[3];
#pragma unroll
  for (int l=0;l<3;l++){
    red[f] = n2[l]; __syncthreads();
    for (int st=32; st>0; st>>=1){ if (f<st) red[f]+=red[f+st]; __syncthreads(); }
    rinv[l] = rsqrtf(red[0] * (1.0f/FDIM) + EPSV);
    __syncthreads();
  }
  float shift = c[n*CDIMV + b_off + f];
#pragma unroll
  for (int m=0;m<MDIM;m++){
    int l = deg_of(m);
    float y = xr[m]*rinv[l];
    y = y * (1.0f + c[n*CDIMV + g_off + (p*3+l)*FDIM + f]);
    if (m==0) y += shift;
    xo[base + m*FDIM + f] = y;
  }
}

// ---------- K3: fused q/k/v projections: one pass over x_pre, 3 accumulators ----------
__global__ void k_qkv(const float* __restrict__ xp,
                      const float* __restrict__ Wq, const float* __restrict__ Wk,
                      const float* __restrict__ Wv,
                      float* __restrict__ q, float* __restrict__ kk, float* __restrict__ v){
  int ct = blockIdx.x, nt = blockIdx.y, pm = blockIdx.z;
  int p = pm / 9, m = pm % 9, l = deg_of(m);
  int lane = threadIdx.x, lo = lane&15, hi = lane>>4;
  const float* A = xp + nt*16*ROWF + (p*MDIM+m)*FDIM;
  int woff = (p*3+l)*FDIM*FDIM + ct*16;
  const float* B0 = Wq + woff;
  const float* B1 = Wk + woff;
  const float* B2 = Wv + woff;
  v8f acq = {}, ack = {}, acv = {};
#pragma unroll
  for (int k=0;k<FDIM;k+=4){
    v2f a, b;
    a.x = A[lo*ROWF + k + 2*hi];
    a.y = A[lo*ROWF + k + 2*hi + 1];
    int r0 = (k + 2*hi)*FDIM + lo, r1 = (k + 2*hi + 1)*FDIM + lo;
    b.x = B0[r0]; b.y = B0[r1]; acq = wmma4(a, b, acq);
    b.x = B1[r0]; b.y = B1[r1]; ack = wmma4(a, b, ack);
    b.x = B2[r0]; b.y = B2[r1]; acv = wmma4(a, b, acv);
  }
#pragma unroll
  for (int r=0;r<8;r++){
    int row = hi*8+r;
    int idx = (nt*16+row)*ROWF + (p*MDIM+m)*FDIM + ct*16 + lo;
    q[idx] = acq[r]; kk[idx] = ack[r]; v[idx] = acv[r];
  }
}

// ---------- K4b: per-dst edge count and cutoff sum ----------
__global__ void k_cnt(const int* __restrict__ dst, const float* __restrict__ cut,
                      float* __restrict__ cnt, float* __restrict__ scut){
  int e = blockIdx.x*blockDim.x + threadIdx.x;
  if (e < N_EDGES){
    int d = dst[e];
    atomicAdd(&cnt[d], 1.0f);
    atomicAdd(&scut[d], cut[e]);
  }
}

// ---------- K4: logits[e,h] = <q[dst], k[src]*e_qk>/sqrt(288); e_qk fused; EPB edges/block ----------
__global__ void k_logits(const float* __restrict__ fe, const float* __restrict__ Weqk,
                         const float* __restrict__ q, const float* __restrict__ kk,
                         const int* __restrict__ src, const int* __restrict__ dst,
                         float* __restrict__ logits, unsigned* __restrict__ lmax){
  __shared__ float ws[FDIM*FDIM];
  __shared__ float fes[MDIM*FDIM];
  int e0 = blockIdx.x*EPB, t = threadIdx.x;
  for (int i=t; i<FDIM*FDIM; i+=64) ws[i] = Weqk[i];
  for (int ei=0; ei<EPB; ei++){
    int e = e0 + ei;
    int s = src[e], d = dst[e];
    __syncthreads();            // covers ws load (ei==0) and prior fes readers
#pragma unroll
    for (int m=0;m<MDIM;m++) fes[m*FDIM+t] = fe[(e*MDIM+m)*FDIM + t];
    __syncthreads();
    if (ei+1 < EPB){            // prefetch next edge's gathered rows + edge features
      int e2 = e + 1;
      int s2 = src[e2], d2 = dst[e2];
      __builtin_prefetch(&q[d2*ROWF + t*18], 0, 0);
      __builtin_prefetch(&kk[s2*ROWF + t*18], 0, 0);
      __builtin_prefetch(&fe[(e2*MDIM)*FDIM + t*9], 0, 0);
    }
    float part = 0.0f;
    for (int m=0;m<MDIM;m++){
      float eqk = 0.0f;
#pragma unroll 8
      for (int f=0; f<FDIM; f++) eqk += fes[m*FDIM+f]*ws[f*FDIM + t];
#pragma unroll
      for (int p=0;p<PDIM;p++){
        int off = p*MDIM*FDIM + m*FDIM + t;
        part += q[d*ROWF + off] * (kk[s*ROWF + off] * eqk);
      }
    }
    for (int mask=8; mask; mask>>=1) part += __shfl_xor(part, mask);
    if ((t & 15) == 0){
      int h = t >> 4;
      float lg = part * 0.05892556509887896f; // 1/sqrt(P*M*DH)=1/sqrt(288)
      logits[e*HHEADS + h] = lg;
      atomicMax(&lmax[d*HHEADS + h], f2key(lg));
    }
  }
}

// ---------- K5: w = exp(logit - max)*cutoff; denom segment-sum ----------
__global__ void k_softmax(const float* __restrict__ logits, const unsigned* __restrict__ lmax,
                          const float* __restrict__ cut, const int* __restrict__ dst,
                          float* __restrict__ attw, float* __restrict__ denom){
  int i = blockIdx.x*blockDim.x + threadIdx.x;
  if (i < N_EDGES*HHEADS){
    int e = i / HHEADS, h = i % HHEADS;
    int d = dst[e];
    float lm = key2f(lmax[d*HHEADS + h]);
    float w = expf(logits[i] - lm) * cut[e];
    attw[i] = w;
    atomicAdd(&denom[d*HHEADS + h], w);
  }
}

// ---------- K6: out[dst] += attn * v[src] * e_v ; e_v fused; EPB edges/block ----------
__global__ void k_scatter(const float* __restrict__ fe, const float* __restrict__ Wev,
                          const float* __restrict__ v, const float* __restrict__ attw,
                          const float* __restrict__ denom,
                          const int* __restrict__ src, const int* __restrict__ dst,
                          float* __restrict__ outb){
  __shared__ float ws[FDIM*FDIM];
  __shared__ float fes[MDIM*FDIM];
  int e0 = blockIdx.x*EPB, t = threadIdx.x;
  for (int i=t; i<FDIM*FDIM; i+=64) ws[i] = Wev[i];
  int h = t >> 4;
  for (int ei=0; ei<EPB; ei++){
    int e = e0 + ei;
    int s = src[e], d = dst[e];
    float aw = attw[e*HHEADS + h] / (denom[d*HHEADS + h] + 1e-9f);
    __syncthreads();
#pragma unroll
    for (int m=0;m<MDIM;m++) fes[m*FDIM+t] = fe[(e*MDIM+m)*FDIM + t];
    __syncthreads();
    if (ei+1 < EPB){
      int e2 = e + 1;
      int s2 = src[e2];
      __builtin_prefetch(&v[s2*ROWF + t*18], 0, 0);
      __builtin_prefetch(&fe[(e2*MDIM)*FDIM + t*9], 0, 0);
    }
    for (int m=0;m<MDIM;m++){
      float ev = 0.0f;
#pragma unroll 8
      for (int f=0; f<FDIM; f++) ev += fes[m*FDIM+f]*ws[f*FDIM + t];
#pragma unroll
      for (int p=0;p<PDIM;p++){
        int off = p*MDIM*FDIM + m*FDIM + t;
        atomicAdd(&outb[d*ROWF + off], aw * v[s*ROWF + off] * ev);
      }
    }
  }
}

// ---------- K7: x_mid = f_nodes + a1 * where(mean_cut<1e-5, x_pre, out@Wo), WMMA ----------
__global__ void k_wo_res(const float* __restrict__ outb, const float* __restrict__ Wo,
                         const float* __restrict__ xp, const float* __restrict__ fnodes,
                         const float* __restrict__ c, const float* __restrict__ cnt,
                         const float* __restrict__ scut, float* __restrict__ xo){
  int ct = blockIdx.x, nt = blockIdx.y, pm = blockIdx.z;
  int p = pm/9, m = pm%9, l = deg_of(m);
  int lane = threadIdx.x, lo = lane&15, hi = lane>>4;
  const float* A = outb + nt*16*ROWF + (p*MDIM+m)*FDIM;
  const float* W = Wo + (p*3+l)*FDIM*FDIM + ct*16;
  v8f acc = {};
  acc = tile_gemm<FDIM>(A, ROWF, W, FDIM, acc, lo, hi);
#pragma unroll
  for (int r=0;r<8;r++){
    int row = hi*8+r; int n = nt*16+row; int g = ct*16+lo;
    int idx = n*ROWF + (p*MDIM+m)*FDIM + g;
    float mc = scut[n] / fmaxf(cnt[n], 1.0f);
    float post = (mc < 1e-5f) ? xp[idx] : acc[r];
    float a1v = c[n*CDIMV + 448 + (p*3+l)*FDIM + g];
    xo[idx] = fnodes[idx] + a1v * post;
  }
}

// ---------- K9: gate from h0 = x_pre2[:,0,0,:] @ W1[0,0], WMMA ----------
__global__ void k_gate(const float* __restrict__ xp2, const float* __restrict__ W1,
                       float* __restrict__ gate){
  int ct = blockIdx.x, nt = blockIdx.y;
  int lane = threadIdx.x, lo = lane&15, hi = lane>>4;
  v8f acc = {};
  acc = tile_gemm<FDIM>(xp2 + nt*16*ROWF, ROWF, W1 + ct*16, NFMLP, acc, lo, hi);
#pragma unroll
  for (int r=0;r<8;r++){
    int row = hi*8+r;
    float s = acc[r];
    float g = (fabsf(s) > 1e-4f) ? (geluf(s)/s) : 0.5f;
    gate[(nt*16+row)*NFMLP + ct*16 + lo] = g;
  }
}

// ---------- K10: fused MLP W1 -> gate -> W2 through LDS, gated residual, WMMA ----------
__global__ void k_mlp(const float* __restrict__ xp2, const float* __restrict__ W1,
                      const float* __restrict__ W2, const float* __restrict__ gate,
                      const float* __restrict__ c, float* __restrict__ xo){
  __shared__ float Hs[16*HSLD];
  int nt = blockIdx.x, pm = blockIdx.y;
  int p = pm/9, m = pm%9, l = deg_of(m);
  int wave = threadIdx.x >> 5, lane = threadIdx.x & 31;
  int lo = lane&15, hi = lane>>4;
  const float* A = xp2 + nt*16*ROWF + (p*MDIM+m)*FDIM;
  const float* W1pl = W1 + (p*3+l)*FDIM*NFMLP;
  const float* W2pl = W2 + (p*3+l)*NFMLP*FDIM;
  // stage 1: H(16x256) = A @ W1, gelu-gated, into LDS. 4 waves x 4 col-tiles.
#pragma unroll
  for (int i=0;i<4;i++){
    int jt = wave*4 + i;
    v8f acc = {};
    acc = tile_gemm<FDIM>(A, ROWF, W1pl + jt*16, NFMLP, acc, lo, hi);
#pragma unroll
    for (int r=0;r<8;r++){
      int row = hi*8+r;
      int gcol = jt*16+lo;
      Hs[row*HSLD + gcol] = acc[r] * gate[(nt*16+row)*NFMLP + gcol];
    }
  }
  __syncthreads();
  // stage 2: Y(16x64) = H @ W2, each wave one 16-col tile, K=256.
  v8f acc2 = {};
  acc2 = tile_gemm<NFMLP>(Hs, HSLD, W2pl + wave*16, FDIM, acc2, lo, hi);
#pragma unroll
  for (int r=0;r<8;r++){
    int row = hi*8+r; int n = nt*16+row; int g = wave*16+lo;
    int idx = n*ROWF + (p*MDIM+m)*FDIM + g;
    float a2v = c[n*CDIMV + 1280 + (p*3+l)*FDIM + g];
    xo[idx] += a2v * acc2[r];
  }
}

extern "C" void kernel_launch(void* const* d_in, const int* in_sizes, int n_in,
                              void* d_out, int out_size, void* d_ws, size_t ws_size,
                              hipStream_t stream){
  (void)in_sizes; (void)n_in; (void)out_size; (void)ws_size;
  const float* f_nodes = (const float*)d_in[0];
  const float* f_edges = (const float*)d_in[1];
  const float* f_time  = (const float*)d_in[2];
  const float* cutoff  = (const float*)d_in[3];
  const float* lnts    = (const float*)d_in[4];
  const float* lntb    = (const float*)d_in[5];
  const float* Wc      = (const float*)d_in[6];
  const float* bc      = (const float*)d_in[7];
  const float* Wq      = (const float*)d_in[8];
  const float* Wk      = (const float*)d_in[9];
  const float* Wv      = (const float*)d_in[10];
  const float* Wo      = (const float*)d_in[11];
  const float* Weqk    = (const float*)d_in[12];
  const float* Wev     = (const float*)d_in[13];
  const float* W1      = (const float*)d_in[14];
  const float* W2      = (const float*)d_in[15];
  const int*   src     = (const int*)d_in[16];
  const int*   dst     = (const int*)d_in[17];
  float* xo = (float*)d_out;
  float* ws = (float*)d_ws;

  size_t o = 0;
  float* u      = ws + o; o += (size_t)N_NODES*TDIM;
  float* c      = ws + o; o += (size_t)N_NODES*CDIMV;
  float* x_pre  = ws + o; o += (size_t)N_NODES*ROWF;
  float* q      = ws + o; o += (size_t)N_NODES*ROWF;
  float* k      = ws + o; o += (size_t)N_NODES*ROWF;
  float* v      = ws + o; o += (size_t)N_NODES*ROWF;
  float* outb   = ws + o; size_t zero_start = o; o += (size_t)N_NODES*ROWF;
  float* logits = ws + o; o += (size_t)N_EDGES*HHEADS;
  unsigned* lmax = (unsigned*)(ws + o); o += (size_t)N_NODES*HHEADS;
  float* denom  = ws + o; o += (size_t)N_NODES*HHEADS;
  float* attw   = ws + o; o += (size_t)N_EDGES*HHEADS;
  float* cnt    = ws + o; o += (size_t)N_NODES;
  float* scut   = ws + o; o += (size_t)N_NODES;
  size_t zero_end = o;
  float* gate   = ws + o; o += (size_t)N_NODES*NFMLP;

  hipMemsetAsync(outb, 0, (zero_end - zero_start)*sizeof(float), stream);

  k_ln_silu<<<N_NODES, 64, 0, stream>>>(f_time, lnts, lntb, u);
  k_gemm_c<<<dim3(CDIMV/16, N_NODES/16), 32, 0, stream>>>(u, Wc, bc, c);
  k_eqv_mod<<<N_NODES*PDIM, 64, 0, stream>>>(f_nodes, c, x_pre, 0, 384);
  k_qkv<<<dim3(FDIM/16, N_NODES/16, 18), 32, 0, stream>>>(x_pre, Wq, Wk, Wv, q, k, v);
  k_cnt<<<(N_EDGES+255)/256, 256, 0, stream>>>(dst, cutoff, cnt, scut);
  k_logits<<<N_EDGES/EPB, 64, 0, stream>>>(f_edges, Weqk, q, k, src, dst, logits, lmax);
  k_softmax<<<(N_EDGES*HHEADS+255)/256, 256, 0, stream>>>(logits, lmax, cutoff, dst, attw, denom);
  k_scatter<<<N_EDGES/EPB, 64, 0, stream>>>(f_edges, Wev, v, attw, denom, src, dst, outb);
  k_wo_res<<<dim3(FDIM/16, N_NODES/16, 18), 32, 0, stream>>>(outb, Wo, x_pre, f_nodes, c, cnt, scut, xo);
  k_eqv_mod<<<N_NODES*PDIM, 64, 0, stream>>>(xo, c, x_pre, 832, 1216);
  k_gate<<<dim3(NFMLP/16, N_NODES/16), 32, 0, stream>>>(x_pre, W1, gate);
  k_mlp<<<dim3(N_NODES/16, 18), 128, 0, stream>>>(x_pre, W1, W2, gate, c, xo);
}